// LMANet_26362509263547
// MI455X (gfx1250) — compile-verified
//
#include <hip/hip_runtime.h>
#include <hip/hip_bf16.h>

typedef __attribute__((ext_vector_type(2))) float v2f;
typedef __attribute__((ext_vector_type(8))) float v8f;

namespace {
constexpr int Bsz = 4;
constexpr int De  = 64;
constexpr int Do  = 128;
constexpr int T   = 4;
constexpr int H   = 128;
constexpr int W   = 256;
constexpr int P2  = 81;     // 9*9 displacements
constexpr int PT  = P2 * T; // 324
constexpr int HW  = H * W;

__device__ __forceinline__ int clampi(int v, int lo, int hi) {
    return v < lo ? lo : (v > hi ? hi : v);
}

// One 16(M=w) x 16(N=j) x 64(K=c) GEMM tile of the correlation.
// EDGE=false: all addresses in-range, selects fold to straight loads.
template<bool EDGE>
__device__ __forceinline__ void corr_tile(const float* __restrict__ krow,
                                          bool hok, int wj, int half,
                                          const v2f* a, v8f& acc) {
    const bool ok = !EDGE || (hok && (wj >= 0) && (wj < W));
    const float* kb = krow + (EDGE ? clampi(wj, 0, W - 1) : wj);
    #pragma unroll
    for (int kk = 0; kk < 16; ++kk) {
        const int c0 = kk * 4 + 2 * half;
        const float vx = kb[(size_t)c0 * (T * HW)];
        const float vy = kb[(size_t)(c0 + 1) * (T * HW)];
        v2f bb;
        bb.x = ok ? vx : 0.0f;
        bb.y = ok ? vy : 0.0f;
        acc = __builtin_amdgcn_wmma_f32_16x16x4_f32(
                false, a[kk], false, bb, (short)0, acc, false, false);
    }
}

// Both j-tiles for one dy + dump of D tiles into per-wave LDS scratch.
template<bool EDGE>
__device__ __forceinline__ void corr_dy(const float* __restrict__ krow,
                                        bool hok, int w0, int half, int ln,
                                        const v2f* a,
                                        float (*scr)[32]) {
    #pragma unroll
    for (int jt = 0; jt < 2; ++jt) {
        const int wj = w0 - 4 + jt * 16 + ln;
        v8f acc = {};
        corr_tile<EDGE>(krow, hok, wj, half, a, acc);
        const int col = jt * 16 + ln;     // D: VGPR r -> M = r + 8*half, N = ln
        #pragma unroll
        for (int r = 0; r < 8; ++r)
            scr[r + 8 * half][col] = acc[r];
    }
}

// Kernel-2 inner product over the 8 channel-tiles for one dy.
template<bool EDGE>
__device__ __forceinline__ void agg_dy(const float* __restrict__ vbase,
                                       int w0, int half, int ln,
                                       const v2f* a, v8f* accs) {
    #pragma unroll 2
    for (int ct = 0; ct < 8; ++ct) {
        v8f acc = accs[ct];
        const int c = ct * 16 + ln;               // B column (N = ln)
        const float* vrow = vbase + (size_t)c * (T * HW);
        #pragma unroll
        for (int s = 0; s < 6; ++s) {
            v2f bb;
            #pragma unroll
            for (int e = 0; e < 2; ++e) {
                const int  j   = 4 * s + e + 2 * half;
                const int  ww  = w0 - 4 + j;
                const bool ok  = !EDGE || ((ww >= 0) && (ww < W));
                const float v  = vrow[EDGE ? clampi(ww, 0, W - 1) : ww];
                const float vv = ok ? v : 0.0f;
                if (e == 0) bb.x = vv; else bb.y = vv;
            }
            acc = __builtin_amdgcn_wmma_f32_16x16x4_f32(
                    false, a[s], false, bb, (short)0, acc, false, false);
        }
        accs[ct] = acc;
    }
}
} // namespace

// ---------------------------------------------------------------------------
// Kernel 1: local correlation via banded f32 WMMA GEMM + joint softmax -> p
// grid: (W/16, H, B), block: 128 threads (4 waves, wave == t)
// ---------------------------------------------------------------------------
__global__ __launch_bounds__(128)
void lma_corr_softmax_kernel(const float* __restrict__ q,
                             const float* __restrict__ m_in,
                             float* __restrict__ p_out)
{
    __shared__ float pv[16][PT];          // [pixel][t*81 + dy*9 + dx]
    __shared__ float scratch[4][16][32];  // per-wave GEMM D-tile dump
    __shared__ float red[2][16][8];       // softmax reductions

    const int tid  = threadIdx.x;
    const int lane = tid & 31;
    const int wv   = tid >> 5;      // wave id == time frame t
    const int half = lane >> 4;
    const int ln   = lane & 15;

    const int w0 = blockIdx.x * 16;
    const int h  = blockIdx.y;
    const int b  = blockIdx.z;
    const int t  = wv;

    // --- A operand: q tile, 16 pixels x 64 channels, resident in registers.
    v2f a[16];
    {
        const float* qb = q + ((size_t)b * De) * HW + (size_t)h * W + w0 + ln;
        #pragma unroll
        for (int kk = 0; kk < 16; ++kk) {
            const int c0 = kk * 4 + 2 * half;
            a[kk].x = qb[(size_t)c0 * HW];
            a[kk].y = qb[(size_t)(c0 + 1) * HW];
        }
    }

    const float* kframe = m_in + ((size_t)b * De * T + t) * HW;
    const bool xin = (w0 >= 4) && (w0 + 27 < W);   // j-strip inside [0,W)

    for (int dy = 0; dy < 9; ++dy) {
        const int  hh  = h + dy - 4;
        const bool hok = (hh >= 0) && (hh < H);
        const int  hhc = clampi(hh, 0, H - 1);
        const float* krow = kframe + (size_t)hhc * W;

        // prefetch next dy row of this frame into cache
        if (dy < 8) {
            const int hn = clampi(hh + 1, 0, H - 1);
            __builtin_prefetch(kframe + (size_t)hn * W + w0 + lane, 0, 1);
        }

        if (hok && xin)
            corr_dy<false>(krow, true, w0, half, ln, a, scratch[wv]);
        else
            corr_dy<true>(krow, hok, w0, half, ln, a, scratch[wv]);

        // make LDS dump visible before cross-lane band extraction
        asm volatile("s_wait_dscnt 0" ::: "memory");

        // extract diagonal band: p_raw[m, dx] = out2[m, m+dx] / De
        #pragma unroll
        for (int ii = 0; ii < 5; ++ii) {
            const int e = lane + 32 * ii;
            if (e < 144) {
                const int m  = e / 9;
                const int dx = e - m * 9;
                pv[m][t * 81 + dy * 9 + dx] =
                    scratch[wv][m][m + dx] * (1.0f / De);
            }
        }
        asm volatile("s_wait_dscnt 0" ::: "memory");
    }
    __syncthreads();

    // --- joint softmax over 324 (d,t) entries per pixel; 8 threads/pixel.
    const int m = tid & 15;
    const int g = tid >> 4;   // 0..7
    float mx = -3.0e38f;
    for (int idx = g; idx < PT; idx += 8) mx = fmaxf(mx, pv[m][idx]);
    red[0][m][g] = mx;
    __syncthreads();
    float mxa = red[0][m][0];
    #pragma unroll
    for (int i = 1; i < 8; ++i) mxa = fmaxf(mxa, red[0][m][i]);

    float s = 0.0f;
    for (int idx = g; idx < PT; idx += 8) {
        const float ev = __expf(pv[m][idx] - mxa);
        pv[m][idx] = ev;
        s += ev;
    }
    red[1][m][g] = s;
    __syncthreads();
    float sa = 0.0f;
    #pragma unroll
    for (int i = 0; i < 8; ++i) sa += red[1][m][i];
    const float inv = 1.0f / sa;

    for (int idx = g; idx < PT; idx += 8) {
        const int tt = idx / 81;
        const int d  = idx - tt * 81;
        p_out[(((size_t)b * P2 + d) * T + tt) * HW + (size_t)h * W + w0 + m] =
            pv[m][idx] * inv;
    }
}

// ---------------------------------------------------------------------------
// Kernel 2: correlation-transpose aggregation via banded f32 WMMA GEMM -> mem
// grid: (W/16, H, B), block: 128 threads (4 waves, wave == t)
// ---------------------------------------------------------------------------
__global__ __launch_bounds__(128)
void lma_aggregate_kernel(const float* __restrict__ p,
                          const float* __restrict__ m_out,
                          float* __restrict__ mem)
{
    __shared__ float memtile[16][Do];   // 8 KB, cross-wave reduction target

    const int tid  = threadIdx.x;
    const int lane = tid & 31;
    const int wv   = tid >> 5;
    const int half = lane >> 4;
    const int ln   = lane & 15;

    const int w0 = blockIdx.x * 16;
    const int h  = blockIdx.y;
    const int b  = blockIdx.z;
    const int t  = wv;

    for (int i = tid; i < 16 * Do; i += 128) (&memtile[0][0])[i] = 0.0f;
    __syncthreads();

    v8f accs[8];
    #pragma unroll
    for (int ct = 0; ct < 8; ++ct) { v8f z = {}; accs[ct] = z; }

    const int m = ln;
    const float* pbase = p + ((size_t)b * P2 * T + t) * HW
                           + (size_t)h * W + w0 + m;   // + d*(T*HW)
    const float* vframe = m_out + ((size_t)b * Do * T + t) * HW;
    const bool xin = (w0 >= 4) && (w0 + 23 < W);       // B strip inside [0,W)

    for (int dy = 0; dy < 9; ++dy) {
        const int  hh  = h + dy - 4;
        const bool hok = (hh >= 0) && (hh < H);
        if (!hok) continue;   // uniform per block: padded rows contribute zero

        // A operand: banded p, M = pixel (ln), K = j (24 values, 6 K-steps);
        // off-band zeros are structural -> clamped load + select always.
        v2f a[6];
        #pragma unroll
        for (int s = 0; s < 6; ++s) {
            #pragma unroll
            for (int e = 0; e < 2; ++e) {
                const int  j   = 4 * s + e + 2 * half;
                const int  dxs = j - 4 - m;           // valid iff in [-4,4]
                const bool ok  = (dxs >= -4) && (dxs <= 4);
                const int  d   = dy * 9 + clampi(dxs, -4, 4) + 4;
                const float v  = pbase[(size_t)d * (T * HW)];
                const float vv = ok ? v : 0.0f;
                if (e == 0) a[s].x = vv; else a[s].y = vv;
            }
        }

        const float* vbase = vframe + (size_t)hh * W + (w0 - 4);
        if (xin) agg_dy<false>(vbase, w0, half, ln, a, accs);
        else     agg_dy<true>(vframe + (size_t)hh * W + (w0 - 4),
                              w0, half, ln, a, accs);
        (void)vbase;
    }

    // cross-wave reduction over the 4 time frames via LDS float atomics
    #pragma unroll
    for (int ct = 0; ct < 8; ++ct) {
        const int c = ct * 16 + ln;
        #pragma unroll
        for (int r = 0; r < 8; ++r)
            atomicAdd(&memtile[r + 8 * half][c], accs[ct][r]);
    }
    __syncthreads();

    for (int i = tid; i < 16 * Do; i += 128) {
        const int mm = i & 15;
        const int c  = i >> 4;
        mem[(((size_t)b * Do + c) * H + h) * W + w0 + mm] =
            memtile[mm][c] * (1.0f / Do);
    }
}

extern "C" void kernel_launch(void* const* d_in, const int* in_sizes, int n_in,
                              void* d_out, int out_size, void* d_ws, size_t ws_size,
                              hipStream_t stream) {
    (void)in_sizes; (void)n_in; (void)d_ws; (void)ws_size; (void)out_size;
    const float* m_in  = (const float*)d_in[0];  // [B, De, T, H, W]
    const float* m_out = (const float*)d_in[1];  // [B, Do, T, H, W]
    const float* q_in  = (const float*)d_in[2];  // [B, De, H, W]
    // d_in[3] = corr_size (hardcoded 9)

    float* memv = (float*)d_out;                              // [B, Do, H, W]
    float* pv   = (float*)d_out + (size_t)Bsz * Do * H * W;   // [B, P2, T, H, W]

    dim3 grid(W / 16, H, Bsz);
    dim3 block(128);
    lma_corr_softmax_kernel<<<grid, block, 0, stream>>>(q_in, m_in, pv);
    lma_aggregate_kernel<<<grid, block, 0, stream>>>(pv, m_out, memv);
}